// AttentionConv_79242146611369
// MI455X (gfx1250) — compile-verified
//
#include <hip/hip_runtime.h>
#include <hip/hip_bf16.h>
#include <math.h>

typedef __attribute__((ext_vector_type(2))) float v2f;
typedef __attribute__((ext_vector_type(8))) float v8f;

#define B_    4
#define CIN_  256
#define CINH_ 128
#define N_    4096
#define K_    20
#define COUT_ 128
#define TOPK_ 20
#define NK_   (N_ * K_)        // 81920 columns per batch
#define NG_   (N_ / 4)         // 1024 n-groups (4 n's = 80 columns) per batch
#define JT_   5                // 5 16-wide j tiles per n-group
#define LST_  81               // LDS row stride (pad 80 -> 81: bank-conflict free)

// ---------------------------------------------------------------------------
// Kernel 1 (fused): WMMA f32 GEMM (q = Wq*abs_x, kk = Wk*x, v = Wv*x) for one
// n-group of 4 n's (80 columns), then in-block softmax over K=20, out partial,
// coalesced score scatter-add, and k_all/v_all extraction. Only q is persisted
// to global (needed by the post-top-k pass).
//
// Wave layout: 8 waves x 16-o tiles cover COUT=128; each wave register-blocks
// 5 j-tiles (accQ/accK/accV[5] = 120 accumulator VGPRs) so each A (weight)
// load feeds 5 WMMAs.
// A (16x4 o x c): lane<16 -> V0=c0,V1=c0+1 ; lane>=16 -> c0+2,c0+3.
// B (4x16 c x j): lane = column; half selects (c0,c0+1) vs (c0+2,c0+3).
// C/D: VGPR r, lane (col,half) -> element (o0 + r + 8*half, col).
// ---------------------------------------------------------------------------
__global__ __launch_bounds__(256) void k_fused_gemm(
    const float* __restrict__ x, const float* __restrict__ abs_x,
    const float* __restrict__ Wq, const float* __restrict__ Wk,
    const float* __restrict__ Wv, const int* __restrict__ idx,
    float* __restrict__ q_ws, float* __restrict__ score,
    float* __restrict__ out_partial, float* __restrict__ k_all,
    float* __restrict__ v_all)
{
    __shared__ float sP[COUT_ * LST_];   // q*kk products, 128 x 80 (padded)
    __shared__ float sV[COUT_ * LST_];   // v,            128 x 80 (padded)
    __shared__ float sK0[COUT_ * 4];     // kk at k==0 per (o, nn)

    const int lane = threadIdx.x & 31;
    const int wave = threadIdx.x >> 5;
    const int col  = lane & 15;
    const int half = lane >> 4;
    const int o0   = wave * 16;
    const int ng   = blockIdx.x % NG_;
    const int b    = blockIdx.x / NG_;
    const size_t j0 = (size_t)ng * 80;   // 4 n's * K=20 columns

    const float* Xb  = x     + (size_t)b * CIN_  * NK_;
    const float* AXb = abs_x + (size_t)b * CINH_ * NK_;

    v8f accQ[JT_] = {};
    v8f accK[JT_] = {};
    v8f accV[JT_] = {};

    // ---- q accumulation over CINH=128 (32 c-steps x 5 WMMA) ----
    for (int c0 = 0; c0 < CINH_; c0 += 4) {
        const int c = c0 + 2 * half;
        v2f a = {};
        a[0] = Wq[(o0 + col) * CINH_ + c];
        a[1] = Wq[(o0 + col) * CINH_ + c + 1];
        const float* B0 = AXb + (size_t)c * NK_ + j0 + col;
        const float* B1 = B0 + NK_;
        __builtin_prefetch(B0 + 4 * (size_t)NK_, 0, 1);  // next c-step column
        #pragma unroll
        for (int jt = 0; jt < JT_; ++jt) {
            v2f bb = {};
            bb[0] = B0[jt * 16];
            bb[1] = B1[jt * 16];
            accQ[jt] = __builtin_amdgcn_wmma_f32_16x16x4_f32(
                false, a, false, bb, (short)0, accQ[jt], false, false);
        }
    }

    // ---- kk and v accumulation over CIN=256 (64 c-steps x 10 WMMA) ----
    for (int c0 = 0; c0 < CIN_; c0 += 4) {
        const int c = c0 + 2 * half;
        v2f ak = {}, av = {};
        ak[0] = Wk[(o0 + col) * CIN_ + c];
        ak[1] = Wk[(o0 + col) * CIN_ + c + 1];
        av[0] = Wv[(o0 + col) * CIN_ + c];
        av[1] = Wv[(o0 + col) * CIN_ + c + 1];
        const float* B0 = Xb + (size_t)c * NK_ + j0 + col;
        const float* B1 = B0 + NK_;
        __builtin_prefetch(B0 + 4 * (size_t)NK_, 0, 1);  // next c-step column
        #pragma unroll
        for (int jt = 0; jt < JT_; ++jt) {
            v2f bb = {};
            bb[0] = B0[jt * 16];
            bb[1] = B1[jt * 16];
            accK[jt] = __builtin_amdgcn_wmma_f32_16x16x4_f32(
                false, ak, false, bb, (short)0, accK[jt], false, false);
            accV[jt] = __builtin_amdgcn_wmma_f32_16x16x4_f32(
                false, av, false, bb, (short)0, accV[jt], false, false);
        }
    }

    // ---- spill: q -> global, p=q*kk and v -> LDS, kk(k==0) -> LDS ----
    #pragma unroll
    for (int jt = 0; jt < JT_; ++jt) {
        #pragma unroll
        for (int r = 0; r < 8; ++r) {
            const int o  = o0 + r + 8 * half;
            const int jj = jt * 16 + col;
            q_ws[((size_t)b * COUT_ + o) * NK_ + j0 + jj] = accQ[jt][r];
            sP[o * LST_ + jj] = accQ[jt][r] * accK[jt][r];
            sV[o * LST_ + jj] = accV[jt][r];
            if ((jj % 20) == 0) sK0[o * 4 + (jj / 20)] = accK[jt][r];
        }
    }
    __syncthreads();

    // ---- epilogue: 512 (o,n) rows; thread -> (o = rid&127, nn = rid>>7) so a
    //      wave's 32 lanes hit 32 consecutive channels (coalesced atomics) ----
    #pragma unroll
    for (int it = 0; it < 2; ++it) {
        const int rid = it * 256 + threadIdx.x;
        const int o   = rid & 127;
        const int nn  = rid >> 7;
        const int n   = ng * 4 + nn;
        const float* prow = &sP[o * LST_ + nn * 20];
        const float* vrow = &sV[o * LST_ + nn * 20];

        float m = -INFINITY;
        #pragma unroll
        for (int k = 0; k < K_; ++k) m = fmaxf(m, prow[k]);
        float e[K_];
        float ssum = 0.0f;
        #pragma unroll
        for (int k = 0; k < K_; ++k) { e[k] = expf(prow[k] - m); ssum += e[k]; }
        const float inv = 1.0f / ssum;

        const int* idxr = idx + ((size_t)b * N_ + n) * K_;
        float outv = 0.0f;
        #pragma unroll
        for (int k = 0; k < K_; ++k) {
            const float a = e[k] * inv;
            outv += a * vrow[k];
            // score layout (b, n, c): lanes add to consecutive channels
            atomicAdd(&score[((size_t)b * N_ + idxr[k]) * COUT_ + o], a);
        }
        const size_t bon = ((size_t)b * COUT_ + o) * N_ + n;
        out_partial[bon] = outv;
        k_all[bon] = sK0[o * 4 + nn];  // kk[..., 0]
        v_all[bon] = vrow[0];          // v[..., 0]
    }
}

// ---------------------------------------------------------------------------
// Kernel 2: zero the score buffer (required every call; graph-replay safe).
// ---------------------------------------------------------------------------
__global__ __launch_bounds__(256) void k_zero(float* __restrict__ p, size_t n)
{
    size_t i = (size_t)blockIdx.x * blockDim.x + threadIdx.x;
    if (i < n) p[i] = 0.0f;
}

// ---------------------------------------------------------------------------
// Kernel 3: top-20 over N=4096 per (b,o) via 20 argmax passes in LDS, then
// gather k_g/v_g from k_all/v_all. score is (b, n, c); it is only 8.4 MB so
// the strided reads stay L2-resident. Slots are filled in descending-score
// order; k_final pairs slot k with q's k-th column, so order matters.
// ---------------------------------------------------------------------------
__global__ __launch_bounds__(256) void k_topk_gather(
    const float* __restrict__ score, const float* __restrict__ k_all,
    const float* __restrict__ v_all, float* __restrict__ kg,
    float* __restrict__ vg)
{
    __shared__ float sc[N_];
    __shared__ float red[256];
    __shared__ int   redi[256];
    __shared__ int   sel[TOPK_];

    const int tid = threadIdx.x;
    const int o   = blockIdx.x % COUT_;
    const int b   = blockIdx.x / COUT_;
    const size_t bo = (size_t)b * COUT_ + o;

    for (int i = tid; i < N_; i += 256)
        sc[i] = score[((size_t)b * N_ + i) * COUT_ + o];
    __syncthreads();

    for (int t = 0; t < TOPK_; ++t) {
        float best = -INFINITY;
        int besti = 0;
        for (int i = tid; i < N_; i += 256) {
            const float s = sc[i];
            if (s > best) { best = s; besti = i; }
        }
        red[tid] = best;
        redi[tid] = besti;
        __syncthreads();
        for (int s = 128; s > 0; s >>= 1) {
            if (tid < s) {
                if (red[tid + s] > red[tid] ||
                    (red[tid + s] == red[tid] && redi[tid + s] < redi[tid])) {
                    red[tid]  = red[tid + s];
                    redi[tid] = redi[tid + s];
                }
            }
            __syncthreads();
        }
        if (tid == 0) {
            sel[t] = redi[0];
            sc[redi[0]] = -INFINITY;
        }
        __syncthreads();
    }

    if (tid < TOPK_) {
        const int bi = sel[tid];
        kg[bo * TOPK_ + tid] = k_all[bo * N_ + bi];
        vg[bo * TOPK_ + tid] = v_all[bo * N_ + bi];
    }
}

// ---------------------------------------------------------------------------
// Kernel 4: per (b,o,n) second softmax against gathered k_g/v_g; add partial.
// ---------------------------------------------------------------------------
__global__ __launch_bounds__(256) void k_final(
    const float* __restrict__ q_ws, const float* __restrict__ kg,
    const float* __restrict__ vg, const float* __restrict__ out_partial,
    float* __restrict__ out)
{
    const size_t gid = (size_t)blockIdx.x * 256 + threadIdx.x;
    const int    n   = (int)(gid % N_);
    const size_t bo  = gid / N_;
    const size_t base = bo * NK_ + (size_t)n * K_;

    const float* kgr = kg + bo * TOPK_;
    const float* vgr = vg + bo * TOPK_;

    float s[K_];
    float m = -INFINITY;
    #pragma unroll
    for (int k = 0; k < K_; ++k) {
        s[k] = q_ws[base + k] * kgr[k];
        m = fmaxf(m, s[k]);
    }
    float ssum = 0.0f;
    #pragma unroll
    for (int k = 0; k < K_; ++k) { s[k] = expf(s[k] - m); ssum += s[k]; }
    const float inv = 1.0f / ssum;
    float out_all = 0.0f;
    #pragma unroll
    for (int k = 0; k < K_; ++k) out_all += s[k] * inv * vgr[k];

    out[gid] = out_partial[gid] + out_all;
}

// ---------------------------------------------------------------------------
extern "C" void kernel_launch(void* const* d_in, const int* in_sizes, int n_in,
                              void* d_out, int out_size, void* d_ws, size_t ws_size,
                              hipStream_t stream)
{
    const float* x     = (const float*)d_in[0];
    const float* abs_x = (const float*)d_in[1];
    // d_in[2] = deg (unused by the reference computation)
    const float* Wq    = (const float*)d_in[3];
    const float* Wk    = (const float*)d_in[4];
    const float* Wv    = (const float*)d_in[5];
    const int*   idx   = (const int*)d_in[6];
    float*       out   = (float*)d_out;

    const size_t big   = (size_t)B_ * COUT_ * NK_;   // 41,943,040 floats (q)
    const size_t small = (size_t)B_ * COUT_ * N_;    //  2,097,152 floats
    float* q_ws        = (float*)d_ws;
    float* score       = q_ws + big;                 // (b, n, c) layout
    float* out_partial = score + small;
    float* k_all       = out_partial + small;
    float* v_all       = k_all + small;
    float* kg          = v_all + small;
    float* vg          = kg + (size_t)B_ * COUT_ * TOPK_;

    k_zero<<<(unsigned)((small + 255) / 256), 256, 0, stream>>>(score, small);
    k_fused_gemm<<<B_ * NG_, 256, 0, stream>>>(
        x, abs_x, Wq, Wk, Wv, idx, q_ws, score, out_partial, k_all, v_all);
    k_topk_gather<<<B_ * COUT_, 256, 0, stream>>>(score, k_all, v_all, kg, vg);
    k_final<<<(unsigned)(small / 256), 256, 0, stream>>>(q_ws, kg, vg, out_partial, out);
}